// UsmFilter_2894807958245
// MI455X (gfx1250) — compile-verified
//
#include <hip/hip_runtime.h>
#include <hip/hip_bf16.h>

// Unsharp mask, fused separable 25-tap Gaussian blur on gfx1250 (MI455X).
// Both 1D conv passes are expressed as banded matmuls on V_WMMA_F32_16X16X4_F32.

#define RADIUS 12
#define KSIZE  25            // 2*RADIUS+1
#define IMG_H  512
#define IMG_W  512
#define TILE   64
#define IN_ROWS 96           // 88 valid + 8 pad so row-groups are 16-aligned for WMMA
#define IN_S    89           // 88 cols + 1 pad (conflict-free column-strided A loads)
#define HT_ROWS 96
#define HT_S    72           // 64 cols + 8 pad (conflict-free vertical B loads)

typedef __attribute__((ext_vector_type(2))) float v2f;
typedef __attribute__((ext_vector_type(8))) float v8f;

// Normalized 1D Gaussian, k[t] = exp(-0.5*((t-12)/5)^2) / sum  (compile-time,
// derived in double precision; reference normalizes the 1D kernel identically
// since outer(k,k) of a normalized k is the normalized 2D kernel).
__device__ __constant__ float KK[KSIZE] = {
    0.004534563f, 0.007183082f, 0.010932381f, 0.015986249f, 0.022459835f,
    0.030317606f, 0.039319818f, 0.048995506f, 0.058658262f, 0.067473066f,
    0.074569280f, 0.079180383f, 0.080779934f, 0.079180383f, 0.074569280f,
    0.067473066f, 0.058658262f, 0.048995506f, 0.039319818f, 0.030317606f,
    0.022459835f, 0.015986249f, 0.010932381f, 0.007183082f, 0.004534563f
};

__device__ __forceinline__ int refl(int p) {
    // jnp.pad mode="reflect": -1 -> 1, H -> H-2
    p = (p < 0) ? -p : p;
    return (p >= IMG_H) ? (2 * IMG_H - 2 - p) : p;
}

__global__ __launch_bounds__(256)
void UsmFilter_kernel(const float* __restrict__ img,
                      const float* __restrict__ param,
                      float* __restrict__ out)
{
    __shared__ float sIn[IN_ROWS * IN_S];   // reflect-padded input halo tile
    __shared__ float sHt[HT_ROWS * HT_S];   // horizontally blurred intermediate

    const int tid  = threadIdx.x;
    const int lane = tid & 31;
    const int wave = tid >> 5;
    const int m    = lane & 15;             // M (A rows) == N (B/C cols) per lane
    const int khi  = (lane >> 4) << 1;      // K sub-offset: lanes 0-15 -> 0, 16-31 -> 2
    const int half8 = (lane >> 4) << 3;     // C/D row offset for upper lane half

    const int bid   = blockIdx.x;
    const int plane = bid >> 6;             // 32*3 = 96 planes, 64 tiles each
    const int tile  = bid & 63;
    const int by    = (tile >> 3) * TILE;
    const int bx    = (tile & 7) * TILE;
    const long planeOff = (long)plane * (IMG_H * IMG_W);
    const float p = param[plane / 3];       // per-image unsharp amount

    // ---- Per-lane banded-kernel WMMA fragments (shared by A-use and B-use) ----
    // KB[r][c] = KK[c-r] for 0 <= c-r <= 24 (16x40 banded matrix, chunked K=4).
    // A-frag of KB and B-frag of KB^T reduce to the same per-lane values.
    v2f kw[10];
#pragma unroll
    for (int j = 0; j < 10; ++j) {
        int d0 = 4 * j + khi - m;
        int d1 = d0 + 1;
        float w0 = 0.f, w1 = 0.f;
        if (d0 >= 0 && d0 < KSIZE) w0 = KK[d0];
        if (d1 >= 0 && d1 < KSIZE) w1 = KK[d1];
        kw[j].x = w0; kw[j].y = w1;
    }

    // ---- Stage 1: cooperative load of 88x88 reflect-padded halo tile ----
    for (int i = tid; i < 88 * 88; i += 256) {
        int r  = i / 88;
        int cc = i - r * 88;
        int gy = refl(by + r  - RADIUS);
        int gx = refl(bx + cc - RADIUS);
        sIn[r * IN_S + cc] = img[planeOff + gy * IMG_W + gx];
    }
    __syncthreads();

    // ---- Stage 2: horizontal blur, Htile = In_sub(16x40) x KB^T(40x16) ----
    // 24 tiles (6 row-groups x 4 col-groups), 3 per wave. Row-group 5 touches
    // pad rows 88-95 (garbage in, garbage out, never read by stage 3).
#pragma unroll
    for (int tt = 0; tt < 3; ++tt) {
        int t  = wave + tt * 8;
        int r0 = (t >> 2) << 4;
        int x0 = (t & 3) << 4;
        v8f h = {};
#pragma unroll
        for (int j = 0; j < 10; ++j) {
            const float* ap = &sIn[(r0 + m) * IN_S + x0 + 4 * j + khi];
            v2f a; a.x = ap[0]; a.y = ap[1];
            h = __builtin_amdgcn_wmma_f32_16x16x4_f32(
                    false, a, false, kw[j], (short)0, h, false, false);
        }
#pragma unroll
        for (int v = 0; v < 8; ++v)
            sHt[(r0 + v + half8) * HT_S + x0 + m] = h[v];
    }
    __syncthreads();

    // ---- Stage 3: vertical blur, Vtile = KB(16x40) x H(40x16); fuse unsharp ----
    // 16 tiles (4x4), 2 per wave. Max htmp row touched = 48+36+3 = 87 (valid).
#pragma unroll
    for (int tt = 0; tt < 2; ++tt) {
        int t    = wave + tt * 8;
        int ty16 = (t >> 2) << 4;
        int tx16 = (t & 3) << 4;
        v8f acc = {};
#pragma unroll
        for (int j = 0; j < 10; ++j) {
            int row = ty16 + 4 * j + khi;
            v2f b;
            b.x = sHt[row * HT_S + tx16 + m];
            b.y = sHt[(row + 1) * HT_S + tx16 + m];
            acc = __builtin_amdgcn_wmma_f32_16x16x4_f32(
                      false, kw[j], false, b, (short)0, acc, false, false);
        }
#pragma unroll
        for (int v = 0; v < 8; ++v) {
            int ly = ty16 + v + half8;
            int lx = tx16 + m;
            float iv = sIn[(ly + RADIUS) * IN_S + lx + RADIUS];
            float o  = (iv - acc[v]) * p + iv;
            out[planeOff + (long)(by + ly) * IMG_W + (bx + lx)] = o;
        }
    }
}

extern "C" void kernel_launch(void* const* d_in, const int* in_sizes, int n_in,
                              void* d_out, int out_size, void* d_ws, size_t ws_size,
                              hipStream_t stream) {
    const float* img   = (const float*)d_in[0];   // [32,3,512,512] fp32
    const float* param = (const float*)d_in[1];   // [32,1] fp32
    float* out = (float*)d_out;                   // [32,3,512,512] fp32

    // 96 planes x (8x8) 64x64 tiles = 6144 blocks, 256 threads (8 wave32)
    UsmFilter_kernel<<<6144, 256, 0, stream>>>(img, param, out);
}